// ChamferLoss_15719580303946
// MI455X (gfx1250) — compile-verified
//
#include <hip/hip_runtime.h>
#include <hip/hip_bf16.h>

typedef float v2f __attribute__((ext_vector_type(2)));
typedef float v8f __attribute__((ext_vector_type(8)));

// ---------------------------------------------------------------------------
// Pack each point (x,y,z) -> (x,y,z,|p|^2) as float4 so the WMMA B operand is
// a single aligned b64 load per lane (lanes 0-15 read (x,y) = K0,K1; lanes
// 16-31 read (z,|q|^2) = K2,K3 at +8 bytes).
// ---------------------------------------------------------------------------
__global__ __launch_bounds__(256) void pack4_kernel(const float* __restrict__ src,
                                                    float* __restrict__ dst, int npts)
{
    int i = blockIdx.x * 256 + threadIdx.x;
    if (i < npts) {
        float x = src[3 * i + 0];
        float y = src[3 * i + 1];
        float z = src[3 * i + 2];
        float4 v = make_float4(x, y, z, x * x + y * y + z * z);
        ((float4*)dst)[i] = v;
    }
}

// ---------------------------------------------------------------------------
// Fast path: one wave owns one 16-row P tile, scans packed Q tiles.
//   A[m][k] = (-2px,-2py,-2pz, 1), B[k][n] = (qx,qy,qz,|q|^2), C[m][n]=|p_m|^2
//   => D = full 16x16 squared-distance tile per V_WMMA_F32_16X16X4_F32.
// ---------------------------------------------------------------------------
__global__ __launch_bounds__(32) void chamfer_dir_packed_kernel(
    const float* __restrict__ P, const float* __restrict__ Q4,
    int N, int M, float* __restrict__ accum)
{
    const int lane = threadIdx.x;            // 0..31, wave32
    const int tilesPerB = N >> 4;
    const int b    = blockIdx.x / tilesPerB;
    const int tile = blockIdx.x - b * tilesPerB;

    const float* Pb = P + (size_t)b * N * 3;

    const int  row_in_tile = lane & 15;
    const int  m  = (tile << 4) + row_in_tile;
    const bool hi = lane >= 16;

    // ---- Build A (loop-invariant) ----
    float px = Pb[3 * m + 0];
    float py = Pb[3 * m + 1];
    float pz = Pb[3 * m + 2];
    float sp = px * px + py * py + pz * pz;   // |p_m|^2

    v2f A;
    A.x = hi ? (-2.0f * pz) : (-2.0f * px);  // K=2 : K=0
    A.y = hi ? 1.0f         : (-2.0f * py);  // K=3 : K=1

    // ---- C: element r holds |p_{tile*16 + r + 8*hi}|^2 ----
    v8f C;
    {
        const int base = hi ? 8 : 0;
#pragma unroll
        for (int r = 0; r < 8; ++r)
            C[r] = __shfl(sp, base + r, 32);
    }

    v8f acc;
#pragma unroll
    for (int r = 0; r < 8; ++r) acc[r] = 3.402823466e38f;

    // Per-lane B pointer: float2 at (point n)*16B + hi*8B, advancing 16 points.
    const float2* qptr = (const float2*)(Q4 + (size_t)b * M * 4)
                         + 2 * row_in_tile + (hi ? 1 : 0);
    const int qtiles = M >> 4;

#pragma unroll 4
    for (int qt = 0; qt < qtiles; ++qt) {
        float2 bb = qptr[(size_t)qt * 32];   // 32 float2 = 16 points stride
        v2f Bm;
        Bm.x = bb.x;
        Bm.y = bb.y;

        v8f D = __builtin_amdgcn_wmma_f32_16x16x4_f32(
            false, A, false, Bm, (short)0, C, false, false);

#pragma unroll
        for (int r = 0; r < 8; ++r) acc[r] = fminf(acc[r], D[r]);
    }

    // ---- Column-min across the 16 lanes of each half, then row sum ----
    float s = 0.0f;
#pragma unroll
    for (int r = 0; r < 8; ++r) {
        float v = acc[r];
        v = fminf(v, __shfl_xor(v, 1, 32));
        v = fminf(v, __shfl_xor(v, 2, 32));
        v = fminf(v, __shfl_xor(v, 4, 32));
        v = fminf(v, __shfl_xor(v, 8, 32));
        s += v;
    }
    s += __shfl_xor(s, 16, 32);
    if (lane == 0) atomicAdd(accum, s);
}

// ---------------------------------------------------------------------------
// Fallback (no packed scratch): build B on the fly (round-1 kernel).
// ---------------------------------------------------------------------------
__global__ __launch_bounds__(32) void chamfer_dir_kernel(
    const float* __restrict__ P, const float* __restrict__ Q,
    int N, int M, float* __restrict__ accum)
{
    const int lane = threadIdx.x;
    const int tilesPerB = N >> 4;
    const int b    = blockIdx.x / tilesPerB;
    const int tile = blockIdx.x - b * tilesPerB;

    const float* Pb = P + (size_t)b * N * 3;
    const float* Qb = Q + (size_t)b * M * 3;

    const int  row_in_tile = lane & 15;
    const int  m  = (tile << 4) + row_in_tile;
    const bool hi = lane >= 16;

    float px = Pb[3 * m + 0], py = Pb[3 * m + 1], pz = Pb[3 * m + 2];
    float sp = px * px + py * py + pz * pz;

    v2f A;
    A.x = hi ? (-2.0f * pz) : (-2.0f * px);
    A.y = hi ? 1.0f         : (-2.0f * py);

    v8f C;
    {
        const int base = hi ? 8 : 0;
#pragma unroll
        for (int r = 0; r < 8; ++r) C[r] = __shfl(sp, base + r, 32);
    }

    v8f acc;
#pragma unroll
    for (int r = 0; r < 8; ++r) acc[r] = 3.402823466e38f;

    const int qtiles = M >> 4;
#pragma unroll 2
    for (int qt = 0; qt < qtiles; ++qt) {
        const int n = (qt << 4) + row_in_tile;
        float qx = Qb[3 * n + 0], qy = Qb[3 * n + 1], qz = Qb[3 * n + 2];
        float sq = qx * qx + qy * qy + qz * qz;
        v2f Bm;
        Bm.x = hi ? qz : qx;
        Bm.y = hi ? sq : qy;
        v8f D = __builtin_amdgcn_wmma_f32_16x16x4_f32(
            false, A, false, Bm, (short)0, C, false, false);
#pragma unroll
        for (int r = 0; r < 8; ++r) acc[r] = fminf(acc[r], D[r]);
    }

    float s = 0.0f;
#pragma unroll
    for (int r = 0; r < 8; ++r) {
        float v = acc[r];
        v = fminf(v, __shfl_xor(v, 1, 32));
        v = fminf(v, __shfl_xor(v, 2, 32));
        v = fminf(v, __shfl_xor(v, 4, 32));
        v = fminf(v, __shfl_xor(v, 8, 32));
        s += v;
    }
    s += __shfl_xor(s, 16, 32);
    if (lane == 0) atomicAdd(accum, s);
}

__global__ void chamfer_zero_kernel(float* ws)
{
    if (threadIdx.x < 2) ws[threadIdx.x] = 0.0f;
}

__global__ void chamfer_finalize_kernel(const float* __restrict__ ws,
                                        float* __restrict__ out,
                                        float invF, float invN)
{
    out[0] = 0.7f * ws[0] * invF + 0.3f * ws[1] * invN;
}

extern "C" void kernel_launch(void* const* d_in, const int* in_sizes, int n_in,
                              void* d_out, int out_size, void* d_ws, size_t ws_size,
                              hipStream_t stream)
{
    const float* pred_f  = (const float*)d_in[0];
    const float* gt_f    = (const float*)d_in[1];
    const float* pred_nf = (const float*)d_in[2];
    const float* gt_nf   = (const float*)d_in[3];
    float* out = (float*)d_out;
    float* ws  = (float*)d_ws;   // ws[0]=filtered sum, ws[1]=nonfiltered sum

    const int B  = 4;
    const int NF = in_sizes[0] / (B * 3);   // 4096
    const int NN = in_sizes[2] / (B * 3);   // 8192
    const int nptsF = B * NF;
    const int nptsN = B * NN;

    // Packed-scratch layout (floats): [0..3] accumulators/pad, then 4 float4 arrays
    const size_t off_pf = 4;
    const size_t off_gf = off_pf + (size_t)nptsF * 4;
    const size_t off_pn = off_gf + (size_t)nptsF * 4;
    const size_t off_gn = off_pn + (size_t)nptsN * 4;
    const size_t need_bytes = (off_gn + (size_t)nptsN * 4) * sizeof(float);

    chamfer_zero_kernel<<<1, 32, 0, stream>>>(ws);

    const float invF = 1.0f / (float)(B * NF);
    const float invN = 1.0f / (float)(B * NN);

    if (ws_size >= need_bytes) {
        // ---- Fast path: pack Q-side operands once, then WMMA scans ----
        pack4_kernel<<<(nptsF + 255) / 256, 256, 0, stream>>>(pred_f,  ws + off_pf, nptsF);
        pack4_kernel<<<(nptsF + 255) / 256, 256, 0, stream>>>(gt_f,    ws + off_gf, nptsF);
        pack4_kernel<<<(nptsN + 255) / 256, 256, 0, stream>>>(pred_nf, ws + off_pn, nptsN);
        pack4_kernel<<<(nptsN + 255) / 256, 256, 0, stream>>>(gt_nf,   ws + off_gn, nptsN);

        dim3 gridF(B * (NF / 16));
        chamfer_dir_packed_kernel<<<gridF, 32, 0, stream>>>(pred_f, ws + off_gf, NF, NF, ws + 0);
        chamfer_dir_packed_kernel<<<gridF, 32, 0, stream>>>(gt_f,   ws + off_pf, NF, NF, ws + 0);

        dim3 gridN(B * (NN / 16));
        chamfer_dir_packed_kernel<<<gridN, 32, 0, stream>>>(pred_nf, ws + off_gn, NN, NN, ws + 1);
        chamfer_dir_packed_kernel<<<gridN, 32, 0, stream>>>(gt_nf,   ws + off_pn, NN, NN, ws + 1);
    } else {
        // ---- Fallback: compute B operand on the fly ----
        dim3 gridF(B * (NF / 16));
        chamfer_dir_kernel<<<gridF, 32, 0, stream>>>(pred_f, gt_f, NF, NF, ws + 0);
        chamfer_dir_kernel<<<gridF, 32, 0, stream>>>(gt_f, pred_f, NF, NF, ws + 0);

        dim3 gridN(B * (NN / 16));
        chamfer_dir_kernel<<<gridN, 32, 0, stream>>>(pred_nf, gt_nf, NN, NN, ws + 1);
        chamfer_dir_kernel<<<gridN, 32, 0, stream>>>(gt_nf, pred_nf, NN, NN, ws + 1);
    }

    chamfer_finalize_kernel<<<1, 1, 0, stream>>>(ws, out, invF, invN);
}